// XMem_11716670783841
// MI455X (gfx1250) — compile-verified
//
#include <hip/hip_runtime.h>
#include <math.h>

// ---------------------------------------------------------------------------
// Problem constants (from the reference)
// ---------------------------------------------------------------------------
#define H_     32
#define W_     56
#define HW_    1792          // H*W
#define CK_    64
#define CV_    512
#define N_     28672         // 16*H*W
#define TOPK_  30
#define KDIM_  128           // 2*CK (fused [mk | mk^2] GEMM)
#define NWAVE_ 16            // waves per workgroup in the sim/topk kernel
#define BLKA_  512           // 16 waves * 32 lanes (wave32)

typedef __attribute__((ext_vector_type(2))) float v2f;
typedef __attribute__((ext_vector_type(8))) float v8f;

// Exact f32 WMMA: D(16x16,f32) = A(16x4,f32) x B(4x16,f32) + C
// 8-arg form: (neg_a, A, neg_b, B, c_mod, C, reuse_a, reuse_b)
__device__ __forceinline__ v8f wmma_f32_16x16x4(v2f a, v2f b, v8f c) {
  return __builtin_amdgcn_wmma_f32_16x16x4_f32(false, a, false, b,
                                               (short)0, c, false, false);
}

// ---------------------------------------------------------------------------
// Kernel A: fused similarity GEMM (f32 WMMA) + streaming top-30 + softmax.
// Grid: 112 workgroups, each owns 16 hw columns. 16 waves split N; each wave
// computes 16x16 sim tiles with 32 x V_WMMA_F32_16X16X4_F32 (K=128).
// Emits per-(hw,k) (index, softmax weight) pairs to workspace.
// ---------------------------------------------------------------------------
__global__ __launch_bounds__(BLKA_) void sim_topk_kernel(
    const float* __restrict__ qk,   // [CK][HW]
    const float* __restrict__ qs,   // [CK][HW]
    const float* __restrict__ mk,   // [CK][N]
    const float* __restrict__ shr,  // [N]
    int*  __restrict__ wsI,         // [HW][TOPK]
    float* __restrict__ wsW)        // [HW][TOPK]
{
  __shared__ float Bls[KDIM_][16];          // fused B operand
  __shared__ float bsq[16];
  __shared__ float lvals[NWAVE_][16][TOPK_];
  __shared__ int   lidxs[NWAVE_][16][TOPK_];
  __shared__ float lmin [NWAVE_][16];
  __shared__ int   lminp[NWAVE_][16];
  __shared__ float topv[16][TOPK_];
  __shared__ int   topi[16][TOPK_];

  const int t       = threadIdx.x;
  const int wave    = t >> 5;
  const int lane    = t & 31;
  const int hf      = lane >> 4;            // lane half (WMMA fragment layout)
  const int lc      = lane & 15;
  const int colBase = blockIdx.x * 16;

  // ---- build fused B: rows [0,64) = 2*qk*qs ; rows [64,128) = -qs ----
  for (int e = t; e < CK_ * 16; e += BLKA_) {
    const int ck = e >> 4, c = e & 15;
    const float kv = qk[ck * HW_ + colBase + c];
    const float sv = qs[ck * HW_ + colBase + c];
    Bls[ck][c]       = 2.f * kv * sv;
    Bls[ck + CK_][c] = -sv;
  }
  // deterministic per-column b_sq = sum_ck qs*qk^2 (no atomics)
  if (t < 16) {
    float s = 0.f;
    for (int ck = 0; ck < CK_; ++ck) {
      const float kv = qk[ck * HW_ + colBase + t];
      const float sv = qs[ck * HW_ + colBase + t];
      s += sv * kv * kv;
    }
    bsq[t] = s;
  }
  for (int e = t; e < NWAVE_ * 16 * TOPK_; e += BLKA_) {
    (&lvals[0][0][0])[e] = -INFINITY;
    (&lidxs[0][0][0])[e] = 0;
  }
  if (t < NWAVE_ * 16) {
    (&lmin[0][0])[t]  = -INFINITY;
    (&lminp[0][0])[t] = 0;
  }
  __syncthreads();

  // ---- B fragments (held in VGPRs for the whole scan) ----
  // B(4x16) layout: VGPR0 = {K=4t | K=4t+2}, VGPR1 = {K=4t+1 | K=4t+3}
  v2f bfrag[32];
#pragma unroll
  for (int tt = 0; tt < 32; ++tt) {
    const int kb = 4 * tt + 2 * hf;
    bfrag[tt].x = Bls[kb][lc];
    bfrag[tt].y = Bls[kb + 1][lc];
  }
  const float bsqc = bsq[lc];

  // ---- stream over N: 112 iterations of 16 rows per wave ----
  for (int tile = 0; tile < N_ / (16 * NWAVE_); ++tile) {
    const int r0 = tile * (16 * NWAVE_) + wave * 16;
    const int r  = r0 + lc;

    // A(16x4) fragments: lanes 0-15 {K,K+1}, lanes 16-31 {K+2,K+3}
    v2f af[16];
#pragma unroll
    for (int tt = 0; tt < 16; ++tt) {
      const int kb = 4 * tt + 2 * hf;
      af[tt].x = mk[(size_t)kb * N_ + r];
      af[tt].y = mk[(size_t)(kb + 1) * N_ + r];
    }

    v8f acc = {0.f, 0.f, 0.f, 0.f, 0.f, 0.f, 0.f, 0.f};
#pragma unroll
    for (int tt = 0; tt < 16; ++tt)          // +2ab term (A = mk)
      acc = wmma_f32_16x16x4(af[tt], bfrag[tt], acc);
#pragma unroll
    for (int tt = 0; tt < 16; ++tt) {        // -a_sq term (A = mk^2)
      v2f a2; a2.x = af[tt].x * af[tt].x; a2.y = af[tt].y * af[tt].y;
      acc = wmma_f32_16x16x4(a2, bfrag[16 + tt], acc);
    }

    // C layout: VGPR j => M=j (lanes 0-15) / M=j+8 (lanes 16-31), N=lane%16
#pragma unroll
    for (int j = 0; j < 8; ++j) {
      const int   row = r0 + j + 8 * hf;
      const float sv  = (acc[j] - bsqc) * shr[row] * 0.125f;  // /sqrt(64)
      // two phases: each half-wave has distinct columns -> conflict-free
#pragma unroll
      for (int ph = 0; ph < 2; ++ph) {
        if (hf == ph && sv > lmin[wave][lc]) {
          const int slot = lminp[wave][lc];
          lvals[wave][lc][slot] = sv;
          lidxs[wave][lc][slot] = row;
          float mn = lvals[wave][lc][0]; int mp = 0;
          for (int s = 1; s < TOPK_; ++s) {
            const float x = lvals[wave][lc][s];
            if (x < mn) { mn = x; mp = s; }
          }
          lmin[wave][lc]  = mn;
          lminp[wave][lc] = mp;
        }
        __builtin_amdgcn_wave_barrier();
        asm volatile("s_wait_dscnt 0x0" ::: "memory");
      }
    }
  }
  __syncthreads();

  // ---- merge 16 per-wave lists (480 candidates) -> top-30, per column ----
  {
    const int c   = wave;        // one column per wave
    const int sub = lane;        // 32 lanes x 15 candidates = 480
    float cv_[15];
#pragma unroll
    for (int q = 0; q < 15; ++q) {
      const int cand = sub * 15 + q;
      cv_[q] = lvals[cand / TOPK_][c][cand % TOPK_];
    }
    for (int kk = 0; kk < TOPK_; ++kk) {
      float bv = -INFINITY; int bc = 0x7fffffff;
#pragma unroll
      for (int q = 0; q < 15; ++q) {
        const int cand = sub * 15 + q;
        if (cv_[q] > bv || (cv_[q] == bv && cand < bc)) { bv = cv_[q]; bc = cand; }
      }
#pragma unroll
      for (int off = 16; off >= 1; off >>= 1) {
        const float ov = __shfl_xor(bv, off);
        const int   oc = __shfl_xor(bc, off);
        if (ov > bv || (ov == bv && oc < bc)) { bv = ov; bc = oc; }
      }
#pragma unroll
      for (int q = 0; q < 15; ++q)
        if (bc == sub * 15 + q) cv_[q] = -INFINITY;   // winner consumed
      if (sub == 0) {
        topv[c][kk] = bv;
        topi[c][kk] = lidxs[bc / TOPK_][c][bc % TOPK_];
      }
    }
    __builtin_amdgcn_wave_barrier();
    asm volatile("s_wait_dscnt 0x0" ::: "memory");

    // softmax over the 30 selected (max-subtracted; identical result)
    const float mx = topv[c][0];
    float e = (lane < TOPK_) ? __expf(topv[c][lane] - mx) : 0.f;
    float ssum = e;
#pragma unroll
    for (int off = 16; off >= 1; off >>= 1) ssum += __shfl_xor(ssum, off);
    const float inv = 1.f / ssum;
    if (lane < TOPK_) {
      const int hw = colBase + c;
      wsW[hw * TOPK_ + lane] = e * inv;
      wsI[hw * TOPK_ + lane] = topi[c][lane];
    }
  }
}

// ---------------------------------------------------------------------------
// Kernel B: tiled transpose mem_value [1024][N] -> [N][1024] so the sparse
// column gather becomes contiguous 4KB reads. Grid (N/32, 1024/32).
// ---------------------------------------------------------------------------
__global__ __launch_bounds__(256) void transpose_kernel(
    const float* __restrict__ mv, float* __restrict__ mt)
{
  __shared__ float tile[32][33];
  const int n0 = blockIdx.x * 32;
  const int j0 = blockIdx.y * 32;
  const int lx = threadIdx.x & 31;
  const int ly = threadIdx.x >> 5;
#pragma unroll
  for (int i = 0; i < 32; i += 8)
    tile[ly + i][lx] = mv[(size_t)(j0 + ly + i) * N_ + n0 + lx];
  __syncthreads();
#pragma unroll
  for (int i = 0; i < 32; i += 8)
    mt[(size_t)(n0 + ly + i) * 1024 + j0 + lx] = tile[lx][ly + i];
}

// ---------------------------------------------------------------------------
// Kernel C: out[j][hw] = sum_k w[hw][k] * value(j, idx[hw][k]); j = b*CV+cv.
// Grid (16 j-blocks of 64, 112 hw-tiles of 16). Output staged via LDS so
// global stores are 64B-coalesced across hw.
// ---------------------------------------------------------------------------
__global__ __launch_bounds__(256) void gather_kernel(
    const float* __restrict__ src,   // transposed [N][1024] or raw [1024][N]
    const int*  __restrict__ wsI,
    const float* __restrict__ wsW,
    float* __restrict__ out,         // [1024][HW]
    int transposed)
{
  __shared__ int   sIdx[16][TOPK_];
  __shared__ float sW  [16][TOPK_];
  __shared__ float otile[64][17];
  const int jb = blockIdx.x;        // 0..15  (64 j each)
  const int ht = blockIdx.y;        // 0..111 (16 hw each)
  const int t  = threadIdx.x;
  const int hwBase = ht * 16;

  for (int e = t; e < 16 * TOPK_; e += 256) {
    const int c = e / TOPK_, k = e % TOPK_;
    sIdx[c][k] = wsI[(hwBase + c) * TOPK_ + k];
    sW[c][k]   = wsW[(hwBase + c) * TOPK_ + k];
  }
  __syncthreads();

  const int c  = t >> 4;            // column within tile
  const int jl = t & 15;            // j covers jl + 16*i, i=0..3
  float acc[4] = {0.f, 0.f, 0.f, 0.f};
  if (transposed) {
    for (int k = 0; k < TOPK_; ++k) {
      const float w = sW[c][k];
      const float* p = src + (size_t)sIdx[c][k] * 1024 + jb * 64 + jl;
      acc[0] += w * p[0];  acc[1] += w * p[16];
      acc[2] += w * p[32]; acc[3] += w * p[48];
    }
  } else {
    for (int k = 0; k < TOPK_; ++k) {
      const float w = sW[c][k];
      const float* p = src + (size_t)(jb * 64 + jl) * N_ + sIdx[c][k];
      acc[0] += w * p[0];
      acc[1] += w * p[(size_t)16 * N_];
      acc[2] += w * p[(size_t)32 * N_];
      acc[3] += w * p[(size_t)48 * N_];
    }
  }
#pragma unroll
  for (int i = 0; i < 4; ++i) otile[jl + 16 * i][c] = acc[i];
  __syncthreads();

  const int cc = t & 15, jr = t >> 4;
#pragma unroll
  for (int i = 0; i < 4; ++i) {
    const int jj = jr + 16 * i;
    out[(size_t)(jb * 64 + jj) * HW_ + hwBase + cc] = otile[jj][cc];
  }
}

// ---------------------------------------------------------------------------
// Launch
// ---------------------------------------------------------------------------
extern "C" void kernel_launch(void* const* d_in, const int* in_sizes, int n_in,
                              void* d_out, int out_size, void* d_ws, size_t ws_size,
                              hipStream_t stream) {
  (void)in_sizes; (void)n_in; (void)out_size;
  const float* qk  = (const float*)d_in[0];   // q_key        [1,CK,H,W]
  const float* qs  = (const float*)d_in[1];   // q_selection  [1,CK,H,W]
  const float* mk  = (const float*)d_in[2];   // mem_key      [1,CK,N]
  const float* shr = (const float*)d_in[3];   // mem_shrinkage[1,N,1]
  const float* mv  = (const float*)d_in[4];   // mem_value    [2,CV,N]
  float* out = (float*)d_out;                 // [2,CV,H,W]

  const size_t IDX_BYTES = (size_t)HW_ * TOPK_ * 4;          // 215,040
  const size_t OFF_W     = IDX_BYTES;
  const size_t OFF_MT    = ((2 * IDX_BYTES) + 4095) & ~(size_t)4095;
  const size_t MT_BYTES  = (size_t)N_ * 1024 * 4;            // ~117.4 MB

  char* ws = (char*)d_ws;
  int*   wsI = (int*)ws;
  float* wsW = (float*)(ws + OFF_W);
  float* mt  = (float*)(ws + OFF_MT);
  const bool useT = (ws_size >= OFF_MT + MT_BYTES);

  sim_topk_kernel<<<HW_ / 16, BLKA_, 0, stream>>>(qk, qs, mk, shr, wsI, wsW);
  if (useT)
    transpose_kernel<<<dim3(N_ / 32, 1024 / 32), 256, 0, stream>>>(mv, mt);
  gather_kernel<<<dim3(16, HW_ / 16), 256, 0, stream>>>(
      useT ? mt : mv, wsI, wsW, out, useT ? 1 : 0);
}